// GCNEncoder_68779606278783
// MI455X (gfx1250) — compile-verified
//
#include <hip/hip_runtime.h>
#include <hip/hip_bf16.h>

typedef __attribute__((ext_vector_type(2))) float v2f;
typedef __attribute__((ext_vector_type(8))) float v8f;

#if defined(__has_builtin)
#if __has_builtin(__builtin_amdgcn_global_load_async_to_lds_b32)
#define HAVE_ASYNC_LDS 1
#endif
#endif
#ifndef HAVE_ASYNC_LDS
#define HAVE_ASYNC_LDS 0
#endif

typedef __attribute__((address_space(1))) int g_int;
typedef __attribute__((address_space(3))) int l_int;

static __device__ __forceinline__ void wait_async0() {
#if defined(__has_builtin)
#if __has_builtin(__builtin_amdgcn_s_wait_asynccnt)
  __builtin_amdgcn_s_wait_asynccnt(0);
#else
  asm volatile("s_wait_asynccnt 0x0" ::: "memory");
#endif
#else
  asm volatile("s_wait_asynccnt 0x0" ::: "memory");
#endif
}

// ---------------------------------------------------------------------------
// Degree / normalization kernels
// ---------------------------------------------------------------------------
__global__ void gcn_init_deg(float* __restrict__ deg, int n) {
  int i = blockIdx.x * blockDim.x + threadIdx.x;
  if (i < n) deg[i] = 1.0f;  // self-loop contributes 1
}

__global__ void gcn_count_deg(const long long* __restrict__ dst,
                              float* __restrict__ deg, long long E) {
  long long e = (long long)blockIdx.x * blockDim.x + threadIdx.x;
  if (e < E) atomicAdd(&deg[(int)dst[e]], 1.0f);
}

__global__ void gcn_finish_dinv(float* __restrict__ deg, int n) {
  int i = blockIdx.x * blockDim.x + threadIdx.x;
  if (i < n) {
    float d = deg[i];
    deg[i] = (d > 0.0f) ? rsqrtf(d) : 0.0f;  // in place: deg -> dinv
  }
}

// ---------------------------------------------------------------------------
// Tiled GEMM, C[M,N] = op(A)[M,K] * B[K,N], exact f32 via v_wmma_f32_16x16x4_f32
// Block: 256 threads = 8 waves. Tile: 128(M) x 64(N), K staged 32 at a time.
// Wave w computes rows [w*16, w*16+16) of the tile, 4 accumulators of 16x16.
// B staged in "paired" LDS layout so each fragment is one aligned ds_load_b64;
// fully in-range B tiles staged via global_load_async_to_lds_b32 (ASYNCcnt).
// ---------------------------------------------------------------------------
template <bool RELU_IN>
__global__ __launch_bounds__(256) void gcn_gemm_wmma(
    const float* __restrict__ A, const float* __restrict__ B,
    float* __restrict__ C, int M, int K, int N) {
  constexpr int MT = 128, NT = 64, KT = 32;
  __shared__ float As[MT][KT + 2];          // stride 34: 8B-aligned K pairs
  __shared__ float Bp[(KT / 4) * NT * 4];   // [q][n][kh] -> float2 pairs

  const int tid  = threadIdx.x;
  const int wave = tid >> 5;
  const int lane = tid & 31;
  const int l16  = lane & 15;
  const int kh   = lane >> 4;        // 0 or 1: which K-pair (ISA A/B layout)
  const int koff = kh * 2;
  const int m0 = blockIdx.x * MT;
  const int n0 = blockIdx.y * NT;

  v8f acc[4] = {};

  const int arow = wave * 16 + l16;

  for (int k0 = 0; k0 < K; k0 += KT) {
    // ---- A tile (128x32): scalar fill (handles clamp / K-tail / fused ReLU)
    for (int i = tid; i < MT * KT; i += 256) {
      int r = i >> 5, c = i & (KT - 1);
      int gr = m0 + r; if (gr >= M) gr = M - 1;
      float v = (k0 + c < K) ? A[(size_t)gr * K + (k0 + c)] : 0.0f;
      if (RELU_IN) v = fmaxf(v, 0.0f);
      As[r][c] = v;
    }

    // ---- B tile (32x64) into paired layout:
    //      Bp[((q*NT + c)*2 + p)*2 + e] = B[k0 + 4q + 2p + e][n0 + c]
    const bool fullB = (k0 + KT <= K) && (n0 + NT <= N);
#if HAVE_ASYNC_LDS
    if (fullB) {
      for (int i = tid; i < KT * NT; i += 256) {
        int r = i >> 6, c = i & (NT - 1);
        int q = r >> 2, p = (r >> 1) & 1, e0 = r & 1;
        float* dp = &Bp[(((q * NT) + c) * 2 + p) * 2 + e0];
        __builtin_amdgcn_global_load_async_to_lds_b32(
            (g_int*)(B + (size_t)(k0 + r) * N + (n0 + c)),
            (l_int*)dp, 0, 0);
      }
      wait_async0();
    } else
#endif
    {
      for (int i = tid; i < KT * NT; i += 256) {
        int r = i >> 6, c = i & (NT - 1);
        int gc = n0 + c; if (gc >= N) gc = N - 1;
        float v = (k0 + r < K) ? B[(size_t)(k0 + r) * N + gc] : 0.0f;
        int q = r >> 2, p = (r >> 1) & 1, e0 = r & 1;
        Bp[(((q * NT) + c) * 2 + p) * 2 + e0] = v;
      }
    }

    // Prefetch a slice of the next A K-tile into cache (global_prefetch_b8).
    if (k0 + KT < K) {
      int r = tid >> 5, c = tid & (KT - 1);
      int gr = m0 + r; if (gr >= M) gr = M - 1;
      __builtin_prefetch(A + (size_t)gr * K + (k0 + KT + c), 0, 1);
    }
    __syncthreads();

#pragma unroll
    for (int kk = 0; kk < KT; kk += 4) {
      v2f a = *(const v2f*)&As[arow][kk + koff];   // one ds_load_b64
      const int q = kk >> 2;
#pragma unroll
      for (int nt = 0; nt < 4; ++nt) {
        int n = nt * 16 + l16;
        v2f b = *(const v2f*)&Bp[(((q * NT) + n) * 2 + kh) * 2];  // ds_load_b64
        acc[nt] = __builtin_amdgcn_wmma_f32_16x16x4_f32(
            false, a, false, b, (short)0, acc[nt], false, false);
      }
    }
    __syncthreads();
  }

  // D layout: VGPR v -> M = v + 8*(lane/16), N = lane&15 (ISA 7.12.2).
  const int crow = m0 + wave * 16 + (kh * 8);
#pragma unroll
  for (int nt = 0; nt < 4; ++nt) {
    int gn = n0 + nt * 16 + l16;
    if (gn < N) {
#pragma unroll
      for (int v = 0; v < 8; ++v) {
        int gm = crow + v;
        if (gm < M) C[(size_t)gm * N + gn] = acc[nt][v];
      }
    }
  }
}

// ---------------------------------------------------------------------------
// out[i,c] = bias[c] + h[i,c] * dinv[i]^2   (self-loop term + bias init)
// ---------------------------------------------------------------------------
__global__ void gcn_selfloop_bias(const float* __restrict__ h,
                                  const float* __restrict__ dinv,
                                  const float* __restrict__ bias,
                                  float* __restrict__ out, int C, int n) {
  long long idx = (long long)blockIdx.x * blockDim.x + threadIdx.x;
  if (idx >= (long long)n * C) return;
  int node = (int)(idx / C);
  int c = (int)(idx % C);
  float di = dinv[node];
  out[idx] = bias[c] + h[idx] * di * di;
}

// ---------------------------------------------------------------------------
// One block per edge: out[dst] += h[src] * dinv[src]*dinv[dst]
// ---------------------------------------------------------------------------
__global__ __launch_bounds__(128) void gcn_edge_scatter(
    const float* __restrict__ h, const long long* __restrict__ src,
    const long long* __restrict__ dst, const float* __restrict__ dinv,
    float* __restrict__ out, int C) {
  long long e = blockIdx.x;
  int s = (int)src[e];
  int d = (int)dst[e];
  float w = dinv[s] * dinv[d];
  const float* hr = h + (size_t)s * C;
  float* orow = out + (size_t)d * C;
  for (int c = threadIdx.x; c < C; c += blockDim.x)
    atomicAdd(&orow[c], hr[c] * w);
}

// ---------------------------------------------------------------------------
extern "C" void kernel_launch(void* const* d_in, const int* in_sizes, int n_in,
                              void* d_out, int out_size, void* d_ws, size_t ws_size,
                              hipStream_t stream) {
  const float*     x   = (const float*)d_in[0];
  const long long* ei  = (const long long*)d_in[1];  // int64 [2, E]
  const float*     W1  = (const float*)d_in[2];
  const float*     b1  = (const float*)d_in[3];
  const float*     W2  = (const float*)d_in[4];
  const float*     b2  = (const float*)d_in[5];
  float*           out = (float*)d_out;

  const int h1    = in_sizes[3];               // 400
  const int h2    = in_sizes[5];               // 800
  const int in_ch = in_sizes[2] / h1;          // 512
  const int n     = in_sizes[0] / in_ch;       // 50000
  const long long E = in_sizes[1] / 2;         // 1.6M
  const long long* srcv = ei;
  const long long* dstv = ei + E;

  // Workspace layout (~240.2 MB):
  //   dinv  @ 0                       : n floats
  //   h1/h2 @ off0  (h2 reuses h1)    : n*h2 floats
  //   a1    @ off1                    : n*h1 floats
  char* ws = (char*)d_ws;
  size_t off0 = ((size_t)n * 4 + 255) & ~(size_t)255;
  size_t off1 = off0 + (((size_t)n * h2 * 4 + 255) & ~(size_t)255);
  float* dinv  = (float*)ws;
  float* h1buf = (float*)(ws + off0);
  float* h2buf = h1buf;
  float* a1    = (float*)(ws + off1);

  // --- degrees -> dinv ---
  gcn_init_deg<<<(n + 255) / 256, 256, 0, stream>>>(dinv, n);
  gcn_count_deg<<<(int)((E + 255) / 256), 256, 0, stream>>>(dstv, dinv, E);
  gcn_finish_dinv<<<(n + 255) / 256, 256, 0, stream>>>(dinv, n);

  // --- layer 1: h1 = x @ W1 ; a1 = scatter(h1) + b1 ---
  dim3 g1((n + 127) / 128, (h1 + 63) / 64);
  gcn_gemm_wmma<false><<<g1, 256, 0, stream>>>(x, W1, h1buf, n, in_ch, h1);
  gcn_selfloop_bias<<<(int)(((long long)n * h1 + 255) / 256), 256, 0, stream>>>(
      h1buf, dinv, b1, a1, h1, n);
  gcn_edge_scatter<<<(int)E, 128, 0, stream>>>(h1buf, srcv, dstv, dinv, a1, h1);

  // --- layer 2: h2 = relu(a1) @ W2 ; out = scatter(h2) + b2 ---
  dim3 g2((n + 127) / 128, (h2 + 63) / 64);
  gcn_gemm_wmma<true><<<g2, 256, 0, stream>>>(a1, W2, h2buf, n, h1, h2);
  gcn_selfloop_bias<<<(int)(((long long)n * h2 + 255) / 256), 256, 0, stream>>>(
      h2buf, dinv, b2, out, h2, n);
  gcn_edge_scatter<<<(int)E, 128, 0, stream>>>(h2buf, srcv, dstv, dinv, out, h2);
}